// BPR_16518444220424
// MI455X (gfx1250) — compile-verified
//
#include <hip/hip_runtime.h>
#include <hip/hip_bf16.h>

typedef __attribute__((ext_vector_type(2))) float v2f;
typedef __attribute__((ext_vector_type(8))) float v8f;

#define HAS_WMMA_F32_4 __has_builtin(__builtin_amdgcn_wmma_f32_16x16x4_f32)

// ---------------------------------------------------------------------------
// dst[r, :64] = dvec[r] * src[r, :64]   (the "+ h*d" term, written first; the
// edge kernel then atomically accumulates SpMM messages on top).
// One wave handles 16 rows x 64 cols. diag(d) @ H done as 4 accumulated
// rank-4 V_WMMA_F32_16X16X4_F32 products — bit-identical to v_mul_f32 RNE
// (each output element gets exactly one nonzero product; other adds are +0).
// ---------------------------------------------------------------------------
__global__ void diag_scale_kernel(const float* __restrict__ src, long srcStride,
                                  const float* __restrict__ dvec,
                                  float* __restrict__ dst, long dstStride,
                                  int numRowBlocks) {
  const int wave = threadIdx.x >> 5;
  const int lane = threadIdx.x & 31;
  const int rb = blockIdx.x * 8 + wave;
  if (rb >= numRowBlocks) return;          // wave-uniform: EXEC stays all-ones
  const long row0 = (long)rb * 16;
  const int half = lane >> 4;              // 0: lanes 0-15, 1: lanes 16-31
  const int lm   = lane & 15;

#if HAS_WMMA_F32_4
  // A-matrix (16x4 f32, 2 VGPRs): lane holds A[m=lm, k=v+2*half].
  // Slab i of diag(d): A_i[m,k] = d[m] iff m == 4*i + k.
  const float dval = dvec[row0 + lm];
  v2f a[4];
#pragma unroll
  for (int i = 0; i < 4; ++i) {
    const int kbase = 4 * i + 2 * half;
    a[i].x = (lm == kbase)     ? dval : 0.0f;
    a[i].y = (lm == kbase + 1) ? dval : 0.0f;
  }

#pragma unroll
  for (int j = 0; j < 4; ++j) {            // four 16-col feature tiles (FACTOR=64)
    v8f c = {0.f, 0.f, 0.f, 0.f, 0.f, 0.f, 0.f, 0.f};
#pragma unroll
    for (int i = 0; i < 4; ++i) {
      // B-matrix (4x16 f32, 2 VGPRs): lane holds B[k=v+2*half, n=lm].
      // B_i[k,n] = src[row0 + 4*i + k, 16*j + n]  (64B coalesced per half-wave)
      const float* bp = src + (row0 + 4 * i + 2 * half) * srcStride + 16 * j + lm;
      v2f b;
      b.x = bp[0];
      b.y = bp[srcStride];
      c = __builtin_amdgcn_wmma_f32_16x16x4_f32(
          /*neg_a=*/false, a[i], /*neg_b=*/false, b,
          /*c_mod=*/(short)0, c, /*reuse_a=*/false, /*reuse_b=*/false);
    }
    // C/D layout: lane holds D[m = v + 8*half, n = lm]
    float* op = dst + (row0 + 8 * half) * dstStride + 16 * j + lm;
#pragma unroll
    for (int v = 0; v < 8; ++v) op[v * dstStride] = c[v];
  }
#else
  // Scalar fallback (same tiling, plain VALU)
#pragma unroll
  for (int j = 0; j < 4; ++j)
#pragma unroll
    for (int v = 0; v < 8; ++v) {
      const long r = row0 + v + 8 * half;
      dst[r * dstStride + 16 * j + lm] = dvec[r] * src[r * srcStride + 16 * j + lm];
    }
#endif
}

// ---------------------------------------------------------------------------
// dstFeat[dstIdx[e], :] += w[e] * srcFeat[srcIdx[e], :]
// 16 lanes per edge, one float4 (16B) per lane -> 256B coalesced gather per
// half-wave; 4x global_atomic_add_f32 per lane resolve in the 192MB L2
// (gene+herb tables are L2-resident). Edge metadata is streamed with
// global_prefetch_b8 hints.
// ---------------------------------------------------------------------------
__global__ void edge_spmm_atomic(const int* __restrict__ srcIdx,
                                 const int* __restrict__ dstIdx,
                                 const float* __restrict__ w,
                                 const float* __restrict__ srcFeat, long srcStride,
                                 float* __restrict__ dstFeat, long dstStride,
                                 int nEdges) {
  const long long t = (long long)blockIdx.x * blockDim.x + threadIdx.x;
  const long long e = t >> 4;
  if (e >= nEdges) return;
  const int c = ((int)t & 15) << 2;

  // speculative prefetch of edge metadata ~4K edges ahead (OOB is dropped)
  __builtin_prefetch(srcIdx + e + 4096, 0, 1);
  __builtin_prefetch(dstIdx + e + 4096, 0, 1);
  __builtin_prefetch(w + e + 4096, 0, 1);

  const int   s  = srcIdx[e];
  const int   d  = dstIdx[e];
  const float wv = w[e];

  const float4 v = *(const float4*)(srcFeat + (long)s * srcStride + c);
  float* o = dstFeat + (long)d * dstStride + c;
  atomicAdd(o + 0, wv * v.x);
  atomicAdd(o + 1, wv * v.y);
  atomicAdd(o + 2, wv * v.z);
  atomicAdd(o + 3, wv * v.w);
}

// dst[r, 0:64] = src[r, 0:64]  (layer-0 slice of the concat output)
__global__ void copy_first_slice(const float* __restrict__ src,
                                 float* __restrict__ dst, long rows) {
  const long long t = (long long)blockIdx.x * blockDim.x + threadIdx.x;
  const long long r = t >> 4;
  if (r >= rows) return;
  const int c = ((int)t & 15) << 2;
  *(float4*)(dst + r * 256 + c) = *(const float4*)(src + r * 64 + c);
}

extern "C" void kernel_launch(void* const* d_in, const int* in_sizes, int n_in,
                              void* d_out, int out_size, void* d_ws, size_t ws_size,
                              hipStream_t stream) {
  (void)n_in; (void)out_size; (void)d_ws; (void)ws_size;

  const float* h0   = (const float*)d_in[0];   // herbs_embedding (H x 64)
  const float* g0   = (const float*)d_in[1];   // genes_embedding (G x 64)
  const float* d_i  = (const float*)d_in[2];   // (H)
  const float* d_j  = (const float*)d_in[3];   // (G)
  const int*   eh   = (const int*)d_in[4];     // edge_herb (E)
  const int*   eg   = (const int*)d_in[5];     // edge_gene (E)
  const float* whg  = (const float*)d_in[6];   // w_hg (E)
  const float* wgh  = (const float*)d_in[7];   // w_gh (E)

  const long H = in_sizes[2];                  // 100000 (multiple of 16)
  const long G = in_sizes[3];                  // 500000 (multiple of 16)
  const int  E = in_sizes[4];                  // 4000000

  float* outH = (float*)d_out;                 // (H x 256)
  float* outG = outH + H * 256;                // (G x 256)

  // layer-0 slices of the concatenation
  copy_first_slice<<<(unsigned)((H * 16 + 255) / 256), 256, 0, stream>>>(h0, outH, H);
  copy_first_slice<<<(unsigned)((G * 16 + 255) / 256), 256, 0, stream>>>(g0, outG, G);

  const int hRB = (int)(H / 16), gRB = (int)(G / 16);
  const unsigned hScaleGrid = (unsigned)((hRB + 7) / 8);
  const unsigned gScaleGrid = (unsigned)((gRB + 7) / 8);
  const unsigned edgeGrid   = (unsigned)(((long long)E * 16 + 255) / 256);

  const float* hPrev = h0; long sh = 64;
  const float* gPrev = g0; long sg = 64;

  for (int L = 1; L <= 3; ++L) {
    float* hNew = outH + L * 64;   // row stride 256
    float* gNew = outG + L * 64;

    // init with the diagonal term: hNew = d_i * hPrev ; gNew = d_j * gPrev
    diag_scale_kernel<<<hScaleGrid, 256, 0, stream>>>(hPrev, sh, d_i, hNew, 256, hRB);
    diag_scale_kernel<<<gScaleGrid, 256, 0, stream>>>(gPrev, sg, d_j, gNew, 256, gRB);

    // hNew[edge_herb] += w_hg * gPrev[edge_gene]
    edge_spmm_atomic<<<edgeGrid, 256, 0, stream>>>(eg, eh, whg, gPrev, sg, hNew, 256, E);
    // gNew[edge_gene] += w_gh * hPrev[edge_herb]
    edge_spmm_atomic<<<edgeGrid, 256, 0, stream>>>(eh, eg, wgh, hPrev, sh, gNew, 256, E);

    hPrev = hNew; sh = 256;
    gPrev = gNew; sg = 256;
  }
}